// MultiBoxLoss_81922206204399
// MI455X (gfx1250) — compile-verified
//
#include <hip/hip_runtime.h>
#include <math.h>

// MultiBox loss for MI455X (gfx1250, wave32).
//   d_in[0] predicted_locs   f32 [128, 8732, 4]
//   d_in[1] predicted_scores f32 [128, 8732, 21]
//   d_in[2] boxes            f32 [128, 32, 4]   (xy)
//   d_in[3] labels           i32 [128, 32]
//   d_in[4] priors_cxcy      f32 [8732, 4]
// Output: single f32 scalar.

#define NIMG   128
#define NP     8732
#define NC     21
#define NOBJ   32
#define BLOCK  256
#define RATIO  3
#define NTILES ((NP + BLOCK - 1) / BLOCK)   // 35
#define TILE_F (BLOCK * NC)                 // 5376 floats per tile buffer

typedef __attribute__((ext_vector_type(2))) float v2f;
typedef __attribute__((ext_vector_type(8))) float v8f;
typedef __attribute__((ext_vector_type(4))) int   v4i;

#define AS1 __attribute__((address_space(1)))
#define AS3 __attribute__((address_space(3)))

// ---- CDNA5 async copy: global -> LDS, b128, tracked by ASYNCcnt ----
__device__ __forceinline__ void async_ld_b128(const float* g, float* l) {
#if __has_builtin(__builtin_amdgcn_global_load_async_to_lds_b128)
    __builtin_amdgcn_global_load_async_to_lds_b128(
        (AS1 v4i*)(uintptr_t)g,   // global source (AS1), const stripped via integer
        (AS3 v4i*)l,              // LDS destination (addrspacecast generic -> AS3)
        0, 0);
#else
    unsigned loff = (unsigned)(uintptr_t)(AS3 void*)l;   // 32-bit LDS offset
    asm volatile("global_load_async_to_lds_b128 %0, %1, off"
                 :: "v"(loff), "v"(g) : "memory");
#endif
}

__device__ __forceinline__ void wait_async0() {
#if __has_builtin(__builtin_amdgcn_s_wait_asynccnt)
    __builtin_amdgcn_s_wait_asynccnt(0);
#else
    asm volatile("s_wait_asynccnt 0x0" ::: "memory");
#endif
}

__device__ __forceinline__ float block_reduce_add(float v, float* redf, int tid) {
    redf[tid] = v;
    __syncthreads();
    for (int s = BLOCK / 2; s > 0; s >>= 1) {
        if (tid < s) redf[tid] += redf[tid + s];
        __syncthreads();
    }
    float r = redf[0];
    __syncthreads();
    return r;
}

__global__ __launch_bounds__(BLOCK) void mbloss_per_image(
    const float* __restrict__ ploc,    // [N,P,4]
    const float* __restrict__ pscore,  // [N,P,21]
    const float* __restrict__ boxes,   // [N,32,4] xy
    const int*   __restrict__ labels,  // [N,32]
    const float* __restrict__ priors,  // [P,4] cxcy
    float* __restrict__ ws_npos,
    float* __restrict__ ws_cepos,
    float* __restrict__ ws_loc,
    float* __restrict__ ws_hard)
{
    const int img = blockIdx.x;
    const int tid = threadIdx.x;

    __shared__ float bx1[NOBJ], by1[NOBJ], bx2[NOBJ], by2[NOBJ];
    __shared__ float bcx[NOBJ], bcy[NOBJ], bww[NOBJ], bhh[NOBJ], barea[NOBJ];
    __shared__ int   lab[NOBJ];
    __shared__ unsigned long long objbest[NOBJ];  // packed (iou_bits<<32)|(~prior)
    __shared__ int   pfo[NOBJ];                   // prior_for_obj
    __shared__ float vals[NP];                    // best IoU per prior; later neg-CE
    __shared__ unsigned char objs[NP];            // best object per prior
    __shared__ float sbuf[2][TILE_F];             // double-buffered score tiles
    __shared__ unsigned int hist[256];
    __shared__ float redf[BLOCK];
    __shared__ int   sh_bin, sh_krem;

    // ---- load per-image objects into LDS ----
    if (tid < NOBJ) {
        const float* b = boxes + ((size_t)img * NOBJ + tid) * 4;
        float x1 = b[0], y1 = b[1], x2 = b[2], y2 = b[3];
        bx1[tid] = x1; by1[tid] = y1; bx2[tid] = x2; by2[tid] = y2;
        bcx[tid] = 0.5f * (x1 + x2);
        bcy[tid] = 0.5f * (y1 + y2);
        bww[tid] = x2 - x1;
        bhh[tid] = y2 - y1;
        barea[tid] = (x2 - x1) * (y2 - y1);
        lab[tid] = labels[(size_t)img * NOBJ + tid];
        objbest[tid] = 0ull;
    }

    // ---- prefetch score tile 0 via async-to-LDS while we do the matching ----
    {
        const int   n128 = (BLOCK * NC) / 4;   // 1344 b128 packets (tile 0 is full)
        const float* g0 = pscore + (size_t)img * NP * NC;
        for (int i = tid; i < n128; i += BLOCK)
            async_ld_b128(g0 + (size_t)i * 4, &sbuf[0][i * 4]);
    }
    __syncthreads();

    // ---- pass 1: IoU matching (prior-major), per-object argmax via ds_max_u64 ----
    for (int p = tid; p < NP; p += BLOCK) {
        const float* pr = priors + (size_t)p * 4;
        float pcx = pr[0], pcy = pr[1], pw = pr[2], ph = pr[3];
        float px1 = pcx - 0.5f * pw, py1 = pcy - 0.5f * ph;
        float px2 = pcx + 0.5f * pw, py2 = pcy + 0.5f * ph;
        float parea = pw * ph;
        float best = -1.0f;
        int bobj = 0;
        #pragma unroll 4
        for (int j = 0; j < NOBJ; ++j) {
            float iw = fminf(bx2[j], px2) - fmaxf(bx1[j], px1);
            float ih = fminf(by2[j], py2) - fmaxf(by1[j], py1);
            iw = fmaxf(iw, 0.0f);
            ih = fmaxf(ih, 0.0f);
            float inter = iw * ih;
            float iou = inter / (barea[j] + parea - inter);
            if (iou > best) { best = iou; bobj = j; }  // strict > : first-max
            unsigned long long key =
                ((unsigned long long)__float_as_uint(iou) << 32) |
                (unsigned long long)(~(unsigned)p);    // ties -> lowest prior wins
            atomicMax(&objbest[j], key);
        }
        vals[p] = best;
        objs[p] = (unsigned char)bobj;
    }
    __syncthreads();

    if (tid < NOBJ) pfo[tid] = (int)(~(unsigned)(objbest[tid] & 0xFFFFFFFFull));
    wait_async0();          // tile 0 resident (per-wave), then publish to all waves
    __syncthreads();

    // ---- pass 2: tiled over priors; async double-buffered score stream ----
    float npos_l = 0.0f, cep_l = 0.0f, loc_l = 0.0f;
    for (int t = 0; t < NTILES; ++t) {
        const int cur = t & 1;

        // kick off tile t+1 into the other buffer (overlaps with compute below)
        if (t + 1 < NTILES) {
            int cnt = NP - (t + 1) * BLOCK;
            if (cnt > BLOCK) cnt = BLOCK;
            const int n128 = (cnt * NC) / 4;   // cnt is a multiple of 4 -> exact
            const float* g = pscore + ((size_t)img * NP + (size_t)(t + 1) * BLOCK) * NC;
            float* l = sbuf[cur ^ 1];
            for (int i = tid; i < n128; i += BLOCK)
                async_ld_b128(g + (size_t)i * 4, l + i * 4);
        }

        const int p = t * BLOCK + tid;
        if (p < NP) {
            int   obj = objs[p];
            float ovl = vals[p];
            #pragma unroll
            for (int j = 0; j < NOBJ; ++j)    // force-match fixup (last write wins)
                if (pfo[j] == p) { obj = j; ovl = 1.0f; }
            int lb = (ovl < 0.5f) ? 0 : lab[obj];

            // cross-entropy over 21 classes, scores read from LDS (stride 21 dwords:
            // coprime with 64 banks -> conflict-free)
            const float* sp = &sbuf[cur][tid * NC];
            float m = sp[0];
            #pragma unroll
            for (int c = 1; c < NC; ++c) m = fmaxf(m, sp[c]);
            float se = 0.0f;
            #pragma unroll
            for (int c = 0; c < NC; ++c) se += expf(sp[c] - m);
            float ce = m + logf(se) - sp[lb];

            __builtin_prefetch(ploc + ((size_t)img * NP + p + BLOCK) * 4, 0, 0);

            if (lb != 0) {
                npos_l += 1.0f;
                cep_l  += ce;
                const float* pr = priors + (size_t)p * 4;
                float pcx = pr[0], pcy = pr[1], pw = pr[2], ph = pr[3];
                float g0 = (bcx[obj] - pcx) * 10.0f / pw;
                float g1 = (bcy[obj] - pcy) * 10.0f / ph;
                float g2 = logf(bww[obj] / pw) * 5.0f;
                float g3 = logf(bhh[obj] / ph) * 5.0f;
                const float* lp = ploc + ((size_t)img * NP + p) * 4;
                float d0 = lp[0] - g0, d1 = lp[1] - g1, d2 = lp[2] - g2, d3 = lp[3] - g3;
                float a0 = fabsf(d0), a1 = fabsf(d1), a2 = fabsf(d2), a3 = fabsf(d3);
                loc_l += (a0 < 1.0f ? 0.5f * d0 * d0 : a0 - 0.5f)
                       + (a1 < 1.0f ? 0.5f * d1 * d1 : a1 - 0.5f)
                       + (a2 < 1.0f ? 0.5f * d2 * d2 : a2 - 0.5f)
                       + (a3 < 1.0f ? 0.5f * d3 * d3 : a3 - 0.5f);
                vals[p] = 0.0f;               // positives contribute 0 to neg pool
            } else {
                vals[p] = ce;
            }
        }

        wait_async0();        // tile t+1 landed (per-wave) ...
        __syncthreads();      // ... and is visible to every wave
    }

    // ---- block reductions (barriers also publish vals[]) ----
    float npos_f = block_reduce_add(npos_l, redf, tid);
    float cep    = block_reduce_add(cep_l,  redf, tid);
    float locs   = block_reduce_add(loc_l,  redf, tid);

    int k = RATIO * (int)npos_f;
    if (k > NP) k = NP;

    // ---- radix-select k-th largest neg-CE (floats >= 0 -> bits ordered) ----
    float hard = 0.0f;
    if (k > 0) {
        unsigned prefix = 0, pmask = 0;
        int krem = k;
        for (int shift = 24; shift >= 0; shift -= 8) {
            hist[tid] = 0;           // BLOCK == 256
            __syncthreads();
            for (int p = tid; p < NP; p += BLOCK) {
                unsigned key = __float_as_uint(vals[p]);
                if ((key & pmask) == prefix)
                    atomicAdd(&hist[(key >> shift) & 255u], 1u);
            }
            __syncthreads();
            if (tid == 0) {
                int cum = 0, b = 255;
                for (; b > 0; --b) {
                    int c = (int)hist[b];
                    if (cum + c >= krem) break;
                    cum += c;
                }
                sh_bin  = b;
                sh_krem = krem - cum;
            }
            __syncthreads();
            prefix |= ((unsigned)sh_bin) << shift;
            pmask  |= 0xFFu << shift;
            krem = sh_krem;
        }
        float T = __uint_as_float(prefix);     // exact k-th largest value
        float sgt = 0.0f;
        for (int p = tid; p < NP; p += BLOCK) {
            float v = vals[p];
            if (v > T) sgt += v;
        }
        sgt = block_reduce_add(sgt, redf, tid);
        hard = sgt + (float)krem * T;          // krem copies of T complete top-k
    }

    if (tid == 0) {
        ws_npos[img]  = npos_f;
        ws_cepos[img] = cep;
        ws_loc[img]   = locs;
        ws_hard[img]  = hard;
    }
}

// 128-way f32 reduction on the matrix pipe: D = ones(16x4) x B(4x16) + C gives
// column sums of B (64 f32 per wave); two accumulating WMMAs cover 128 values.
__device__ __forceinline__ float wmma_sum128(const float* __restrict__ v, int lane) {
    v2f a; a[0] = 1.0f; a[1] = 1.0f;
    v8f c = {};
    v2f b;
    b[0] = v[2 * lane]; b[1] = v[2 * lane + 1];
    c = __builtin_amdgcn_wmma_f32_16x16x4_f32(false, a, false, b, (short)0, c, false, false);
    b[0] = v[64 + 2 * lane]; b[1] = v[64 + 2 * lane + 1];
    c = __builtin_amdgcn_wmma_f32_16x16x4_f32(false, a, false, b, (short)0, c, false, false);
    float s = c[0];
    s += __shfl_xor(s, 1, 32);
    s += __shfl_xor(s, 2, 32);
    s += __shfl_xor(s, 4, 32);
    s += __shfl_xor(s, 8, 32);
    return s;
}

__global__ __launch_bounds__(32) void mbloss_final(
    const float* __restrict__ ws_npos,
    const float* __restrict__ ws_cepos,
    const float* __restrict__ ws_loc,
    const float* __restrict__ ws_hard,
    float* __restrict__ out)
{
    const int lane = threadIdx.x;     // one wave32, EXEC all ones around WMMAs
    float tp = wmma_sum128(ws_npos,  lane);
    float ce = wmma_sum128(ws_cepos, lane);
    float lc = wmma_sum128(ws_loc,   lane);
    float hd = wmma_sum128(ws_hard,  lane);
    if (lane == 0) {
        float conf = (ce + hd) / tp;
        float lloc = lc / (tp * 4.0f);
        out[0] = conf + lloc;         // ALPHA == 1
    }
}

extern "C" void kernel_launch(void* const* d_in, const int* in_sizes, int n_in,
                              void* d_out, int out_size, void* d_ws, size_t ws_size,
                              hipStream_t stream) {
    const float* ploc   = (const float*)d_in[0];
    const float* pscore = (const float*)d_in[1];
    const float* boxes  = (const float*)d_in[2];
    const int*   labels = (const int*)d_in[3];
    const float* priors = (const float*)d_in[4];
    float* out = (float*)d_out;

    float* ws      = (float*)d_ws;   // 4 * 128 floats = 2 KB scratch
    float* ws_npos = ws;
    float* ws_cep  = ws + NIMG;
    float* ws_loc  = ws + 2 * NIMG;
    float* ws_hard = ws + 3 * NIMG;

    mbloss_per_image<<<NIMG, BLOCK, 0, stream>>>(ploc, pscore, boxes, labels, priors,
                                                 ws_npos, ws_cep, ws_loc, ws_hard);
    mbloss_final<<<1, 32, 0, stream>>>(ws_npos, ws_cep, ws_loc, ws_hard, out);
}